// SymCoreLayer_10557029614344
// MI455X (gfx1250) — compile-verified
//
#include <hip/hip_runtime.h>
#include <math.h>

typedef float v2f __attribute__((ext_vector_type(2)));
typedef float v8f __attribute__((ext_vector_type(8)));

#define NSEQ    8
#define S       4096
#define D       128
#define WINDOW  16
#define HALF    8
#define NS      (S - WINDOW + 1)   /* 4081 valid window starts */
#define EPSW    0.01f
#define TINYW   1e-12f
#define THREADS 256
#define NWAVE   (THREADS / 32)
#define NCHUNK  (S / 16)           /* 256 */

static_assert(THREADS == NCHUNK, "replay phase assumes one thread per chunk");

// ---------------------------------------------------------------------------
// Kernel 1: per-sequence. WMMA Gram matrices -> window symmetry flags ->
// parallel chunked transfer-function scan -> gather indices -> (optional) copy.
// ---------------------------------------------------------------------------
__global__ __launch_bounds__(THREADS)
void sym_compress_kernel(const float* __restrict__ x, float* __restrict__ out,
                         int* __restrict__ gws, int use_ws) {
  __shared__ float          gram[NWAVE][768];     // 3 x 16x16 Gram per wave
  __shared__ unsigned char  symA[S];
  __shared__ unsigned char  exitT[NCHUNK][16];
  __shared__ unsigned char  emitT[NCHUNK][16];
  __shared__ unsigned short gidxA[S];
  __shared__ unsigned short EArr[NCHUNK];
  __shared__ int            OArr[NCHUNK];

  const int tid = threadIdx.x;
  const int seq = blockIdx.x;
  const float* xs = x + (size_t)seq * S * D;
  float* os = out + (size_t)seq * S * D;

  // init gather-index sentinel (written later, after a barrier)
  for (int p = tid; p < S; p += THREADS) gidxA[p] = 0xFFFFu;

  const int wid  = tid >> 5;
  const int lane = tid & 31;
  const int lrow = lane & 15;          // matrix row (A) / col (B) for this lane
  const int kh   = (lane >> 4) << 1;   // K-half select: 0 or 2
  float* gw = &gram[wid][0];

  // ---- Phase A: Gram matrices via v_wmma_f32_16x16x4_f32, window flags ----
  for (int b = wid; b < S / 16; b += NWAVE) {
    const int r = b * 16;
    const float* pa = xs + (size_t)(r + lrow) * D + kh;
    int rb = r + 16 + lrow;
    if (rb > S - 1) rb = S - 1;        // clamp OOB rows (values unused)
    const float* pb = xs + (size_t)rb * D + kh;

    v8f c1 = {}; v8f c2 = {}; v8f c3 = {};
    for (int k0 = 0; k0 < D; k0 += 4) {
      v2f fa = *(const v2f*)(pa + k0);   // A-frag == B-frag layout for f32 wmma
      v2f fb = *(const v2f*)(pb + k0);
      c1 = __builtin_amdgcn_wmma_f32_16x16x4_f32(false, fa, false, fa, (short)0, c1, false, false);
      c2 = __builtin_amdgcn_wmma_f32_16x16x4_f32(false, fa, false, fb, (short)0, c2, false, false);
      c3 = __builtin_amdgcn_wmma_f32_16x16x4_f32(false, fb, false, fb, (short)0, c3, false, false);
    }

    // Dump C layout (VGPR i, half-wave hw -> M = i + 8*hw, N = lane&15) to LDS.
    const int mb = (lane >> 4) * 8;
#pragma unroll
    for (int i2 = 0; i2 < 8; ++i2) {
      const int m = mb + i2;
      gw[m * 16 + lrow]       = c1[i2];
      gw[256 + m * 16 + lrow] = c2[i2];
      gw[512 + m * 16 + lrow] = c3[i2];
    }

    // Lanes 0..15 evaluate windows i = r+e from Gram entries (same-wave LDS
    // ordering guarantees visibility; no barrier needed).
    if (lane < 16) {
      const int e = lane;
      float wn2 = 0.f, h11 = 0.f, dotv = 0.f, mirX = 0.f;
#pragma unroll
      for (int k = 0; k < 16; ++k) {
        int rr = e + k;
        float dgn = (rr < 16) ? gw[rr * 16 + rr]
                              : gw[512 + (rr - 16) * 16 + (rr - 16)];
        wn2 += dgn;
        if (k < 8) h11 += dgn;
      }
#pragma unroll
      for (int k = 0; k < 8; ++k) {
        int r1 = e + k;
        int r2 = r1 + 8;                       // lag-8 pair (h1 . h2)
        dotv += (r2 < 16) ? gw[r1 * 16 + r2]
              : (r1 < 16) ? gw[256 + r1 * 16 + (r2 - 16)]
                          : gw[512 + (r1 - 16) * 16 + (r2 - 16)];
        int r3 = e + 15 - k;                   // mirror pair
        mirX += (r3 < 16) ? gw[r1 * 16 + r3]
              : (r1 < 16) ? gw[256 + r1 * 16 + (r3 - 16)]
                          : gw[512 + (r1 - 16) * 16 + (r3 - 16)];
      }
      float h22  = wn2 - h11;
      float per2 = fmaxf(h11 + h22 - 2.f * dotv, 0.f);
      float mir2 = fmaxf(2.f * wn2 - 4.f * mirX, 0.f);
      float sc   = dotv / (h11 + TINYW);
      float scl2 = fmaxf(h22 - 2.f * sc * dotv + sc * sc * h11, 0.f);
      float wn   = sqrtf(fmaxf(wn2, 0.f)) + TINYW;
      float thr  = EPSW * wn;
      int symv = (sqrtf(per2) <= thr) || (sqrtf(mir2) <= thr) || (sqrtf(scl2) <= thr);
      int iw = r + e;
      symA[iw] = (iw < NS) ? (unsigned char)symv : (unsigned char)0;
    }
  }
  __syncthreads();

  // ---- Phase B1: per-chunk transfer functions (entry offset -> exit, emits)
  for (int job = tid; job < NCHUNK * 16; job += THREADS) {
    int c = job >> 4, e = job & 15;
    int j = c * 16 + e;
    int lim = (c + 1) * 16;
    int emits = 0;
    while (j < lim) {
      if (symA[j]) { emits += HALF; j += WINDOW; }
      else         { emits += 1;    j += 1; }
    }
    exitT[c][e] = (unsigned char)(j - lim);    // jumps never skip a chunk
    emitT[c][e] = (unsigned char)emits;
  }
  __syncthreads();

  // ---- Phase B2: serial composition over 256 chunks (cheap) ----
  if (tid == 0) {
    int E = 0, O = 0;
    for (int c = 0; c < NCHUNK; ++c) {
      EArr[c] = (unsigned short)E;
      OArr[c] = O;
      O += (int)emitT[c][E];
      E = (int)exitT[c][E];
    }
  }
  __syncthreads();

  // ---- Phase B3: per-chunk replay writing gather indices ----
  {
    int c = tid;
    int j = c * 16 + (int)EArr[c];
    int off = OArr[c];
    int lim = (c + 1) * 16;
    while (j < lim) {
      if (symA[j]) {
#pragma unroll
        for (int t = 0; t < HALF; ++t) gidxA[off + t] = (unsigned short)(j + t);
        off += HALF; j += WINDOW;
      } else {
        gidxA[off] = (unsigned short)j; off += 1; j += 1;
      }
    }
  }
  __syncthreads();

  // ---- Phase C: emit gather indices to workspace, or gather in-block ----
  if (use_ws) {
    int* g = gws + seq * S;
    for (int p = tid; p < S; p += THREADS) {
      unsigned short v = gidxA[p];
      g[p] = (v == 0xFFFFu) ? -1 : (int)v;
    }
  } else {
    const float4* in4 = (const float4*)xs;
    float4* out4 = (float4*)os;
    for (int idx = tid; idx < S * (D / 4); idx += THREADS) {
      int p = idx >> 5;                 // D/4 == 32
      int q = idx & 31;
      unsigned short g = gidxA[p];
      float4 v;
      if (g != 0xFFFFu) v = in4[(int)g * 32 + q];
      else { v.x = 0.f; v.y = 0.f; v.z = 0.f; v.w = 0.f; }
      out4[idx] = v;
    }
  }
}

// ---------------------------------------------------------------------------
// Kernel 2: full-GPU vectorized gather (used when workspace is available).
// ---------------------------------------------------------------------------
__global__ __launch_bounds__(256)
void sym_gather_kernel(const float* __restrict__ x, const int* __restrict__ gws,
                       float* __restrict__ out) {
  const long total = (long)NSEQ * S * (D / 4);
  const float4* in4 = (const float4*)x;
  float4* out4 = (float4*)out;
  for (long t = (long)blockIdx.x * blockDim.x + threadIdx.x; t < total;
       t += (long)gridDim.x * blockDim.x) {
    long pg = t >> 5;                      // global padded row
    int q  = (int)(t & 31);
    int sq = (int)(pg >> 12);              // S == 4096
    int p  = (int)(pg & (S - 1));
    int g  = gws[sq * S + p];
    float4 v; v.x = 0.f; v.y = 0.f; v.z = 0.f; v.w = 0.f;
    if (g >= 0) v = in4[((long)sq * S + g) * 32 + q];
    out4[t] = v;
  }
}

extern "C" void kernel_launch(void* const* d_in, const int* in_sizes, int n_in,
                              void* d_out, int out_size, void* d_ws, size_t ws_size,
                              hipStream_t stream) {
  (void)in_sizes; (void)n_in; (void)out_size;
  const float* x = (const float*)d_in[0];
  float* out = (float*)d_out;
  const int use_ws =
      (d_ws != nullptr && ws_size >= (size_t)NSEQ * S * sizeof(int)) ? 1 : 0;
  sym_compress_kernel<<<dim3(NSEQ), dim3(THREADS), 0, stream>>>(
      x, out, (int*)d_ws, use_ws);
  if (use_ws) {
    sym_gather_kernel<<<dim3(1024), dim3(256), 0, stream>>>(
        x, (const int*)d_ws, out);
  }
}